// GraphConvolution1_41592463295065
// MI455X (gfx1250) — compile-verified
//
#include <hip/hip_runtime.h>
#include <hip/hip_bf16.h>
#include <stdint.h>

typedef __attribute__((ext_vector_type(2))) float        v2f;
typedef __attribute__((ext_vector_type(8))) float        v8f;
typedef __attribute__((ext_vector_type(4))) unsigned int v4u;
typedef __attribute__((ext_vector_type(8))) int          v8i;
typedef __attribute__((ext_vector_type(4))) int          v4i;

#define F_DIM 512
#define KC    32                     // K-chunk staged in LDS per double-buffer slot
#define TILE_B_FLOATS (KC * 128)     // 32 x 128 f32 tile
#define TILE_B_BYTES  (TILE_B_FLOATS * 4)

#if __has_builtin(__builtin_amdgcn_tensor_load_to_lds)
#define HAVE_TDM 1
#endif

// ---------------------------------------------------------------------------
// Zero-fill d_out (harness poisons it with 0xAA before timing).
// ---------------------------------------------------------------------------
__global__ void zero_f32_kernel(float4* __restrict__ p, long long n4) {
    long long i = (long long)blockIdx.x * blockDim.x + threadIdx.x;
    if (i < n4) {
        p[i] = make_float4(0.f, 0.f, 0.f, 0.f);
    }
}

// ---------------------------------------------------------------------------
// Compiler fence: makes the LDS buffer escape (so the TDM's writes into it
// cannot be "proven dead") and clobbers memory so LDS reads are not folded
// or hoisted across buffer swaps. Emits no instructions.
// ---------------------------------------------------------------------------
__device__ __forceinline__ void lds_publish(const void* p) {
    asm volatile("" : : "v"(p) : "memory");
}

// ---------------------------------------------------------------------------
// TDM helpers: DMA a KC x 128 f32 tile of `weight` (row stride F_DIM) into LDS.
// D# layout per CDNA5 ISA §8 (group0 128b, group1 256b; groups 2/3 zero for 2D).
// Issued by wave 0 only (TDM ignores EXEC; whole-wave branch guards the issue).
// ---------------------------------------------------------------------------
__device__ __forceinline__ void tdm_wait_tensorcnt0() {
#if defined(HAVE_TDM) && __has_builtin(__builtin_amdgcn_s_wait_tensorcnt)
    __builtin_amdgcn_s_wait_tensorcnt(0);
#endif
}

__device__ __forceinline__ void tdm_load_tile_b(const float* __restrict__ gsrc,
                                                unsigned lds_byte_off,
                                                float* __restrict__ lds_dst,
                                                int lane) {
#ifdef HAVE_TDM
    unsigned long long ga = (unsigned long long)(uintptr_t)gsrc;

    v4u g0;
    g0[0] = 1u;                                            // count=1 (valid), user mode
    g0[1] = lds_byte_off;                                  // lds_addr (bytes)
    g0[2] = (unsigned)(ga & 0xffffffffu);                  // global_addr[31:0]
    g0[3] = (unsigned)((ga >> 32) & 0x01ffffffu)           // global_addr[56:32]
            | (2u << 30);                                  // type=2 ("image")

    v8i g1;
    g1[0] = (int)(2u << 16);          // wg_mask=0 (not in cluster), data_size=4B
    g1[1] = (int)(512u << 16);        // tensor_dim0[15:0]=512 (bits 79:48 low half)
    g1[2] = (int)(512u << 16);        // tensor_dim0 hi=0 | tensor_dim1[15:0]=512
    g1[3] = (int)(128u << 16);        // tensor_dim1 hi=0 | tile_dim0=128
    g1[4] = (int)KC;                  // tile_dim1=KC, tile_dim2=0
    g1[5] = (int)F_DIM;               // tensor_dim0_stride = 512 elements
    g1[6] = 0;                        // stride hi | tensor_dim1_stride lo
    g1[7] = 0;

    v4i gz = {0, 0, 0, 0};
#if defined(__clang_major__) && (__clang_major__ >= 23)
    v8i gz8 = {0, 0, 0, 0, 0, 0, 0, 0};
    __builtin_amdgcn_tensor_load_to_lds(g0, g1, gz, gz, gz8, 0);
#else
    __builtin_amdgcn_tensor_load_to_lds(g0, g1, gz, gz, 0);
#endif
    lds_publish(lds_dst);             // tie the DMA to the LDS object
    (void)lane;
#else
    // Fallback (also keeps the host pass clean): wave-0 cooperative copy.
    for (int i = lane; i < TILE_B_FLOATS; i += 32) {
        lds_dst[i] = gsrc[(i >> 7) * F_DIM + (i & 127)];
    }
    (void)lds_byte_off;
#endif
}

// ---------------------------------------------------------------------------
// Stage 1: support = x @ weight  (fp32 WMMA 16x16x4, TDM-staged B tiles)
//
// Block = 256 threads (8 waves) computing a 128-row x 128-col tile.
// Wave w owns rows [row0 + 16w, +16) with 8 v8f accumulators across 128 cols.
// B tiles (KC x 128) are DMA'd into double-buffered LDS by the TDM and read
// back as ds loads (offset pattern merges into ds_load_2addr_b32).
// ---------------------------------------------------------------------------
__global__ void __launch_bounds__(256)
gemm_wmma_f32_kernel(const float* __restrict__ x,
                     const float* __restrict__ w,
                     float* __restrict__ support,
                     int n_row_tiles) {
    __shared__ float bsm[2][TILE_B_FLOATS];                // 2 x 16 KB

    const int lane = threadIdx.x & 31;
    const int m    = lane & 15;
    const int hi   = lane >> 4;
    const int wave = threadIdx.x >> 5;

    const int col0    = blockIdx.y * 128;
    const int rowTile = blockIdx.x * 8 + wave;
    const bool active = rowTile < n_row_tiles;             // whole-wave predicate
    const int row0    = rowTile * 16;

    v8f acc[8];
#pragma unroll
    for (int j = 0; j < 8; ++j) acc[j] = (v8f)(0.f);

    const float* __restrict__ xrow = x + (size_t)(row0 + m) * F_DIM;

    // Preload chunk 0 into buffer 0.
    if (threadIdx.x < 32) {
        tdm_load_tile_b(w + col0, 0u, bsm[0], lane);
        tdm_wait_tensorcnt0();
    }
    __syncthreads();
    lds_publish(bsm);

    int buf = 0;
    const int nchunk = F_DIM / KC;                         // 16
    for (int c = 0; c < nchunk; ++c) {
        // Kick off the next tile's DMA while computing on the current one.
        if (threadIdx.x < 32 && (c + 1) < nchunk) {
            tdm_load_tile_b(w + (size_t)(c + 1) * KC * F_DIM + col0,
                            (unsigned)((buf ^ 1) * TILE_B_BYTES),
                            bsm[buf ^ 1], lane);
        }

        if (active) {
            const float* __restrict__ bt = bsm[buf];
            const int kc = c * KC;
#pragma unroll
            for (int s = 0; s < 8; ++s) {
                // A fragment: contiguous float2 per lane (k = kc+4s+2*hi+{0,1}).
                v2f a = *(const v2f*)(xrow + kc + 4 * s + 2 * hi);
                const float* __restrict__ bp = bt + (4 * s + 2 * hi) * 128 + m;
#pragma unroll
                for (int j = 0; j < 8; ++j) {
                    v2f b;
                    b.x = bp[16 * j];          // row k,   col col0+16j+m
                    b.y = bp[16 * j + 128];    // row k+1, same col
                    acc[j] = __builtin_amdgcn_wmma_f32_16x16x4_f32(
                        /*neg_a=*/false, a,
                        /*neg_b=*/false, b,
                        /*c_mod=*/(short)0, acc[j],
                        /*reuse_a=*/false, /*reuse_b=*/false);
                }
            }
        }

        if (threadIdx.x < 32) tdm_wait_tensorcnt0();       // next tile landed
        __syncthreads();                                   // all waves done w/ buf
        lds_publish(bsm);                                  // no folding across swap
        buf ^= 1;
    }

    if (active) {
        // D layout: VGPR r -> rows (row0 + r + 8*hi), col = col0 + 16*j + m.
#pragma unroll
        for (int j = 0; j < 8; ++j) {
            float* __restrict__ dst =
                support + (size_t)(row0 + 8 * hi) * F_DIM + (col0 + 16 * j + m);
#pragma unroll
            for (int r = 0; r < 8; ++r) {
                dst[(size_t)r * F_DIM] = acc[j][r];
            }
        }
    }
}

// ---------------------------------------------------------------------------
// Stage 2: SpMM scatter:  out[rows[e]] += vals[e] * support[cols[e]]
// One wave32 per edge; float4-coalesced gathers + native f32 global atomics.
// ---------------------------------------------------------------------------
__global__ void __launch_bounds__(256)
spmm_atomic_kernel(const int* __restrict__ rows,
                   const int* __restrict__ cols,
                   const float* __restrict__ vals,
                   const float* __restrict__ support,
                   float* __restrict__ out,
                   int n_edges) {
    const int lane = threadIdx.x & 31;
    const int wave = threadIdx.x >> 5;
    const long long e = (long long)blockIdx.x * (blockDim.x >> 5) + wave;
    if (e >= n_edges) return;

    const int   r = rows[e];
    const int   c = cols[e];
    const float v = vals[e];

    const float4* __restrict__ src =
        (const float4*)(support + (long long)c * F_DIM);
    float* __restrict__ dst = out + (long long)r * F_DIM;

#pragma unroll
    for (int q = 0; q < 4; ++q) {
        const int idx4 = lane + 32 * q;        // 0..127 float4 slots
        float4 s = src[idx4];
        const int base = idx4 * 4;
        atomicAdd(dst + base + 0, v * s.x);
        atomicAdd(dst + base + 1, v * s.y);
        atomicAdd(dst + base + 2, v * s.z);
        atomicAdd(dst + base + 3, v * s.w);
    }
}

// ---------------------------------------------------------------------------
// Launch
// ---------------------------------------------------------------------------
extern "C" void kernel_launch(void* const* d_in, const int* in_sizes, int n_in,
                              void* d_out, int out_size, void* d_ws, size_t ws_size,
                              hipStream_t stream) {
    const float* x       = (const float*)d_in[0];   // [N, 512]
    const float* weight  = (const float*)d_in[1];   // [512, 512]
    const int*   adjRows = (const int*)d_in[2];     // [E]
    const int*   adjCols = (const int*)d_in[3];     // [E]
    const float* adjVals = (const float*)d_in[4];   // [E]

    const int n_nodes = in_sizes[0] / F_DIM;
    const int n_edges = in_sizes[2];

    float* out     = (float*)d_out;
    float* support = (float*)d_ws;                  // [N, 512] scratch

    // 1) zero output accumulator
    {
        const long long n4 = (long long)n_nodes * (F_DIM / 4);
        const int threads = 256;
        const long long blocks = (n4 + threads - 1) / threads;
        zero_f32_kernel<<<(unsigned)blocks, threads, 0, stream>>>(
            (float4*)out, n4);
    }

    // 2) dense GEMM via f32 WMMA with TDM-staged B tiles
    {
        const int n_row_tiles = n_nodes / 16;       // 100000 / 16 = 6250
        dim3 grid((n_row_tiles + 7) / 8, F_DIM / 128);
        gemm_wmma_f32_kernel<<<grid, 256, 0, stream>>>(x, weight, support,
                                                       n_row_tiles);
    }

    // 3) SpMM scatter with f32 atomics (one wave per edge)
    {
        const int threads = 256;                    // 8 waves -> 8 edges/block
        const int edges_per_block = threads / 32;
        const long long blocks =
            ((long long)n_edges + edges_per_block - 1) / edges_per_block;
        spmm_atomic_kernel<<<(unsigned)blocks, threads, 0, stream>>>(
            adjRows, adjCols, adjVals, support, out, n_edges);
    }
}